// MultiBoxLoss_67267777790468
// MI455X (gfx1250) — compile-verified
//
#include <hip/hip_runtime.h>
#include <stdint.h>

// ---------------------------------------------------------------------------
// MultiBoxLoss (SSD) for MI455X / gfx1250.
// Bandwidth-bound (~140MB traffic, ~0.2 GFLOP): no GEMM structure -> no WMMA.
// CDNA5 paths used: async global->LDS copies (ASYNCcnt) + 320KB LDS for a
// full-row in-LDS radix select (hard negative mining).
// ---------------------------------------------------------------------------

#define BB   128
#define PP   8732
#define CC   21
#define TPB  256

// LDS byte offset of a __shared__ object: generic LDS pointers carry the
// workgroup-relative LDS offset in their low 32 bits (ISA 10.2);
// async-to-LDS takes that offset in a VGPR.
__device__ __forceinline__ uint32_t lds_off(const void* p) {
  return (uint32_t)(uintptr_t)p;
}

__device__ __forceinline__ void async_copy_b32(uint32_t lds_addr, const void* gaddr) {
  asm volatile("global_load_async_to_lds_b32 %0, %1, off"
               :: "v"(lds_addr), "v"((uint64_t)(uintptr_t)gaddr) : "memory");
}
__device__ __forceinline__ void async_copy_b128(uint32_t lds_addr, const void* gaddr) {
  asm volatile("global_load_async_to_lds_b128 %0, %1, off"
               :: "v"(lds_addr), "v"((uint64_t)(uintptr_t)gaddr) : "memory");
}
__device__ __forceinline__ void wait_async0() {
#if __has_builtin(__builtin_amdgcn_s_wait_asynccnt)
  __builtin_amdgcn_s_wait_asynccnt(0);
#else
  asm volatile("s_wait_asynccnt 0" ::: "memory");
#endif
}

// Deterministic 256-thread tree reduction.
__device__ __forceinline__ float block_reduce_f(float v, float* s_red) {
  const int tid = threadIdx.x;
  s_red[tid] = v;
  __syncthreads();
  for (int off = TPB / 2; off > 0; off >>= 1) {
    if (tid < off) s_red[tid] += s_red[tid + off];
    __syncthreads();
  }
  float r = s_red[0];
  __syncthreads();
  return r;
}

// ---------------------------------------------------------------------------
// Kernel 1: per-prior cross entropy.  Grid = B*P/TPB = 4366 (exact), one prior
// per thread.  conf tile (TPB*21 floats = 21504B) staged into LDS with the
// async engine (b32: the 84B/prior stride is only 4B-aligned).
// ---------------------------------------------------------------------------
__global__ void __launch_bounds__(TPB)
mbl_ce_kernel(const float* __restrict__ conf, const int* __restrict__ tgt,
              float* __restrict__ ce_out) {
  __shared__ __align__(16) float s_conf[TPB * CC];

  const int tid = threadIdx.x;
  const int64_t base = (int64_t)blockIdx.x * TPB;   // first prior of this block
  const float* src = conf + base * CC;

  const uint32_t l0 = lds_off(s_conf);
  for (int i = tid; i < TPB * CC; i += TPB)
    async_copy_b32(l0 + (uint32_t)i * 4u, src + i);
  wait_async0();
  __syncthreads();

  float v[CC];
#pragma unroll
  for (int j = 0; j < CC; ++j) v[j] = s_conf[tid * CC + j];

  float m = v[0];
#pragma unroll
  for (int j = 1; j < CC; ++j) m = fmaxf(m, v[j]);
  float s = 0.0f;
#pragma unroll
  for (int j = 0; j < CC; ++j) s += __expf(v[j] - m);
  const float lse = m + __logf(s);

  const int t = tgt[base + tid];                    // 0..20
  ce_out[base + tid] = lse - v[t];
}

// ---------------------------------------------------------------------------
// Kernel 2: one block per batch row.  Async-copies the CE row (34928B =
// 2183 x b128, 16B-aligned) into LDS, folds positives (sum_pos_ce, num_pos,
// smooth-L1 on loc), converts the row to lc in place, then exact top-K sum via
// 4-pass radix select on the float bit patterns (all lc >= 0 => monotone).
// ---------------------------------------------------------------------------
__global__ void __launch_bounds__(TPB)
mbl_mine_kernel(const float* __restrict__ ce, const int* __restrict__ tgt,
                const float* __restrict__ loc, const float* __restrict__ loct,
                float* __restrict__ row_ll, float* __restrict__ row_lc,
                int* __restrict__ row_np) {
  __shared__ __align__(16) float s_lc[PP];          // 34928 B
  __shared__ unsigned s_hist[256];
  __shared__ float s_red[TPB];
  __shared__ unsigned s_sel[2];

  const int tid = threadIdx.x;
  const int b = blockIdx.x;
  const float* src = ce + (size_t)b * PP;

  const uint32_t l0 = lds_off(s_lc);
  for (int i = tid; i < PP / 4; i += TPB)           // 2183 chunks of 16B
    async_copy_b128(l0 + (uint32_t)i * 16u, src + i * 4);
  wait_async0();
  __syncthreads();

  // ---- positives pass: sum_pos_ce, num_pos, smooth-L1; zero lc at positives
  float lossl = 0.0f, spce = 0.0f;
  int np = 0;
  for (int p = tid; p < PP; p += TPB) {
    const int t = tgt[(size_t)b * PP + p];
    if (t > 0) {
      spce += s_lc[p];
      ++np;
      s_lc[p] = 0.0f;
      const size_t o = ((size_t)b * PP + p) * 4;
#pragma unroll
      for (int j = 0; j < 4; ++j) {
        const float d = loc[o + j] - loct[o + j];
        const float ad = fabsf(d);
        lossl += (ad < 1.0f) ? 0.5f * d * d : (ad - 0.5f);
      }
    }
  }
  __syncthreads();

  const float spce_t  = block_reduce_f(spce, s_red);
  const float lossl_t = block_reduce_f(lossl, s_red);
  const int   np_t    = (int)block_reduce_f((float)np, s_red);  // <=8732, exact

  const unsigned K = (unsigned)min(3 * np_t, PP - 1);

  // ---- radix select: K-th largest lc value (bit pattern monotone, lc >= 0)
  float sum_top = 0.0f;
  if (K > 0) {
    unsigned prefix = 0, cntG = 0, rem = K;
    for (int shift = 24; shift >= 0; shift -= 8) {
      s_hist[tid] = 0;
      __syncthreads();
      const unsigned hm = (shift == 24) ? 0u : (0xFFFFFFFFu << (shift + 8));
      for (int p = tid; p < PP; p += TPB) {
        const unsigned k = __float_as_uint(s_lc[p]);
        if ((k & hm) == (prefix & hm))
          atomicAdd(&s_hist[(k >> shift) & 255u], 1u);
      }
      __syncthreads();
      if (tid == 0) {
        unsigned cum = 0;
        int v = 255;
        for (; v > 0; --v) {
          const unsigned h = s_hist[v];
          if (cum + h >= rem) break;
          cum += h;
        }
        s_sel[0] = (unsigned)v;
        s_sel[1] = cum;
      }
      __syncthreads();
      const unsigned v = s_sel[0], cum = s_sel[1];
      prefix |= (v << shift);
      rem -= cum;
      cntG += cum;
      __syncthreads();
    }
    // exact top-K sum, tie-insensitive: values > T fully counted, remainder
    // of the quota filled with copies of the threshold value T.
    const float tv = __uint_as_float(prefix);
    float sg = 0.0f;
    for (int p = tid; p < PP; p += TPB) {
      const unsigned k = __float_as_uint(s_lc[p]);
      if (k > prefix) sg += __uint_as_float(k);
    }
    sg = block_reduce_f(sg, s_red);
    sum_top = sg + (float)(K - cntG) * tv;
  }

  if (tid == 0) {
    row_ll[b] = lossl_t;
    row_lc[b] = spce_t + sum_top;   // Σ_pos ce + Σ_(mined neg) ce
    row_np[b] = np_t;
  }
}

// ---------------------------------------------------------------------------
// Kernel 3: final deterministic reduction over the 128 rows.
// ---------------------------------------------------------------------------
__global__ void __launch_bounds__(BB)
mbl_finalize_kernel(const float* __restrict__ rl, const float* __restrict__ rc,
                    const int* __restrict__ rn, float* __restrict__ out) {
  __shared__ float sa[BB], sb[BB];
  __shared__ int sc[BB];
  const int tid = threadIdx.x;
  sa[tid] = rl[tid];
  sb[tid] = rc[tid];
  sc[tid] = rn[tid];
  __syncthreads();
  for (int off = BB / 2; off > 0; off >>= 1) {
    if (tid < off) {
      sa[tid] += sa[tid + off];
      sb[tid] += sb[tid + off];
      sc[tid] += sc[tid + off];
    }
    __syncthreads();
  }
  if (tid == 0) {
    const float N = (float)sc[0];
    out[0] = sa[0] / N;
    out[1] = sb[0] / N;
  }
}

extern "C" void kernel_launch(void* const* d_in, const int* in_sizes, int n_in,
                              void* d_out, int out_size, void* d_ws, size_t ws_size,
                              hipStream_t stream) {
  (void)in_sizes; (void)n_in; (void)out_size; (void)ws_size;
  const float* loc_data  = (const float*)d_in[0];   // [B,P,4] f32
  const float* conf_data = (const float*)d_in[1];   // [B,P,C] f32
  const float* loc_t     = (const float*)d_in[2];   // [B,P,4] f32
  const int*   conf_t    = (const int*)d_in[3];     // [B,P]   i32
  float* out = (float*)d_out;                       // {loss_l/N, loss_c/N}

  // Workspace layout: ce [B*P] f32 (4.47MB) | row_ll[128] | row_lc[128] | row_np[128]
  float* ce_ws = (float*)d_ws;
  float* rl = ce_ws + (size_t)BB * PP;
  float* rc = rl + BB;
  int*   rn = (int*)(rc + BB);

  mbl_ce_kernel<<<dim3((BB * PP) / TPB), dim3(TPB), 0, stream>>>(conf_data, conf_t, ce_ws);
  mbl_mine_kernel<<<dim3(BB), dim3(TPB), 0, stream>>>(ce_ws, conf_t, loc_data, loc_t, rl, rc, rn);
  mbl_finalize_kernel<<<dim3(1), dim3(BB), 0, stream>>>(rl, rc, rn, out);
}